// Interaction_65627100283516
// MI455X (gfx1250) — compile-verified
//
#include <hip/hip_runtime.h>
#include <math.h>

// Problem constants (match reference)
#define Nn 32768
#define Ee 524288
#define Dd 128
#define TD 384
#define CUTF 5.0f
#define GCH 21      // 20 bessel channels + 1 env channel
#define GROW 84     // 4 * 21 floats per node

typedef __attribute__((ext_vector_type(2))) float v2f;
typedef __attribute__((ext_vector_type(8))) float v8f;

__device__ __forceinline__ v8f wmma_f32_k4(v2f a, v2f b, v8f c) {
  // D = A(16x4,f32) * B(4x16,f32) + C(16x16,f32)
  return __builtin_amdgcn_wmma_f32_16x16x4_f32(
      /*neg_a=*/false, a, /*neg_b=*/false, b,
      /*c_mod=*/(short)0, c, /*reuse_a=*/false, /*reuse_b=*/false);
}

// ---------------------------------------------------------------------------
// Kernel 1: per-edge accumulation of per-node sufficient statistics
//   G[n][m][c] : m in {1, ux, uy, uz}; c<20: env*bessel_c*w_m ; c==20: env*w_m
// ---------------------------------------------------------------------------
__global__ __launch_bounds__(256) void edge_accum_kernel(
    const int* __restrict__ nbr_idx,       // neighbour_index flat [2,E]; use row 1
    const float* __restrict__ edge_vectors, // [E,3]
    const float* __restrict__ freq,         // [20]
    float* __restrict__ G)                  // [N,4,21]
{
  int e = blockIdx.x * blockDim.x + threadIdx.x;
  if (e >= Ee) return;
  int n = nbr_idx[Ee + e];

  float x = edge_vectors[3 * e + 0];
  float y = edge_vectors[3 * e + 1];
  float z = edge_vectors[3 * e + 2];
  float d2 = x * x + y * y + z * z;
  float dist = sqrtf(d2);
  float inv = 1.0f / dist;
  float ux = x * inv, uy = y * inv, uz = z * inv;

  // DimeNet polynomial envelope, p = 6
  float t = dist * (1.0f / CUTF);
  float t2 = t * t;
  float t6 = t2 * t2 * t2;
  float env = 1.0f + t6 * (-28.0f + t * (48.0f + t * (-21.0f)));
  env = (dist < CUTF) ? env : 0.0f;

  float w0 = env, w1 = env * ux, w2 = env * uy, w3 = env * uz;

  const float SQ = 0.6324555320336759f;   // sqrt(2/cutoff)
  float bess[20];
#pragma unroll
  for (int r = 0; r < 20; ++r)
    bess[r] = SQ * __sinf(freq[r] * dist) * inv;

  float* Grow = G + (size_t)n * GROW;
#pragma unroll
  for (int r = 0; r < 20; ++r) {
    float b = bess[r];
    unsafeAtomicAdd(&Grow[0 * GCH + r], w0 * b);
    unsafeAtomicAdd(&Grow[1 * GCH + r], w1 * b);
    unsafeAtomicAdd(&Grow[2 * GCH + r], w2 * b);
    unsafeAtomicAdd(&Grow[3 * GCH + r], w3 * b);
  }
  unsafeAtomicAdd(&Grow[0 * GCH + 20], w0);
  unsafeAtomicAdd(&Grow[1 * GCH + 20], w1);
  unsafeAtomicAdd(&Grow[2 * GCH + 20], w2);
  unsafeAtomicAdd(&Grow[3 * GCH + 20], w3);
}

// ---------------------------------------------------------------------------
// Kernel 2: fused node kernel, one workgroup (8 wave32 waves) per 16 nodes.
//   H   = silu(S@W1 + b1)            [16,128]   (WMMA f32 K=4, 32 steps)
//   Phi = H@W2 + b2                  [16,384]
//   F   = G@W_f (bias handled in combine)  [64,384] phased 48 cols at a time
//   delta_s / delta_v combine (exact, no atomics)
// LDS strides padded (132/388/22/52) for bank-conflict avoidance.
// ---------------------------------------------------------------------------
__global__ __launch_bounds__(256) void node_fused_kernel(
    const float* __restrict__ S,    // [N,128] scalar embeddings
    const float* __restrict__ W1,   // [128,128]
    const float* __restrict__ b1,   // [128]
    const float* __restrict__ W2,   // [128,384]
    const float* __restrict__ b2,   // [384]
    const float* __restrict__ Wf,   // [20,384]
    const float* __restrict__ bf,   // [384]
    const float* __restrict__ G,    // [N,4,21]
    const float* __restrict__ vemb, // [N,128,3]
    float* __restrict__ dv,         // [N,128,3]
    float* __restrict__ dsc)        // [N,128]
{
  __shared__ float sS[16 * 132];
  __shared__ float sH[16 * 132];
  __shared__ float sPhi[16 * 388];
  __shared__ float sG[64 * 22];
  __shared__ float sF[64 * 52];

  const int tid = threadIdx.x;
  const int w = tid >> 5;        // wave id 0..7 (wave32)
  const int lane = tid & 31;
  const int half = lane >> 4;    // ISA f32 WMMA lane-half (K split)
  const int r = lane & 15;       // A: M row; B/C: N column
  const int n0 = blockIdx.x * 16;

  // ---- load S tile and G tile into LDS ----
  for (int i = tid; i < 16 * 128; i += 256) {
    int nl = i >> 7, c = i & 127;
    sS[nl * 132 + c] = S[(size_t)(n0 + nl) * 128 + c];
  }
  for (int i = tid; i < 16 * GROW; i += 256) {
    int nl = i / GROW, rem = i - nl * GROW;
    int m = rem / GCH, c = rem - m * GCH;
    sG[(nl * 4 + m) * 22 + c] = G[(size_t)(n0 + nl) * GROW + rem];
  }
  __syncthreads();

  // ---- Stage 1: H = silu(S@W1 + b1); one 16-col tile per wave ----
  {
    const int cb = w * 16;
    v8f acc = {};
    for (int kk = 0; kk < 32; ++kk) {
      const int krow = kk * 4 + half * 2;
      v2f a, b;
      a.x = sS[r * 132 + krow];
      a.y = sS[r * 132 + krow + 1];
      b.x = W1[krow * 128 + cb + r];
      b.y = W1[(krow + 1) * 128 + cb + r];
      acc = wmma_f32_k4(a, b, acc);
    }
    const float bias = b1[cb + r];
#pragma unroll
    for (int i = 0; i < 8; ++i) {
      const int row = half * 8 + i;   // C layout: VGPR i -> M=i (lo) / M=8+i (hi)
      float v = acc[i] + bias;
      sH[row * 132 + cb + r] = v * (1.0f / (1.0f + __expf(-v)));  // silu
    }
  }
  __syncthreads();

  // ---- Stage 2: Phi = H@W2 + b2; three 16-col tiles per wave ----
#pragma unroll
  for (int t = 0; t < 3; ++t) {
    const int cb = (w + t * 8) * 16;
    v8f acc = {};
    for (int kk = 0; kk < 32; ++kk) {
      const int krow = kk * 4 + half * 2;
      v2f a, b;
      a.x = sH[r * 132 + krow];
      a.y = sH[r * 132 + krow + 1];
      b.x = W2[krow * 384 + cb + r];
      b.y = W2[(krow + 1) * 384 + cb + r];
      acc = wmma_f32_k4(a, b, acc);
    }
    const float bias = b2[cb + r];
#pragma unroll
    for (int i = 0; i < 8; ++i)
      sPhi[(half * 8 + i) * 388 + cb + r] = acc[i] + bias;
  }
  __syncthreads();

  // ---- Stage 3: phase over d in chunks of 16; F = G@Wf for needed cols ----
  for (int p = 0; p < 8; ++p) {
    // 12 F tiles: row-tile m (channel), col-select jsel in {0,1,2} -> j = jsel*128 + p*16
    for (int tt = w; tt < 12; tt += 8) {   // wave-uniform -> EXEC stays full
      const int m = tt & 3, jsel = tt >> 2;
      const int jg = jsel * 128 + p * 16;
      v8f acc = {};
#pragma unroll
      for (int kk = 0; kk < 5; ++kk) {     // K = 20 exactly
        const int krow = kk * 4 + half * 2;
        v2f a, b;
        a.x = sG[(r * 4 + m) * 22 + krow];
        a.y = sG[(r * 4 + m) * 22 + krow + 1];
        b.x = Wf[krow * 384 + jg + r];
        b.y = Wf[(krow + 1) * 384 + jg + r];
        acc = wmma_f32_k4(a, b, acc);
      }
#pragma unroll
      for (int i = 0; i < 8; ++i)
        sF[(m * 16 + half * 8 + i) * 52 + jsel * 16 + r] = acc[i];
    }
    __syncthreads();

    // ---- combine: 256 threads == 16 nodes x 16 d values ----
    {
      const int nl = tid >> 4, dd = tid & 15;
      const int d = p * 16 + dd;
      const int n = n0 + nl;
      const float gE0 = sG[(nl * 4 + 0) * 22 + 20];
      const float gE1 = sG[(nl * 4 + 1) * 22 + 20];
      const float gE2 = sG[(nl * 4 + 2) * 22 + 20];
      const float gE3 = sG[(nl * 4 + 3) * 22 + 20];
      const float bfa = bf[d], bfb = bf[128 + d], bfc = bf[256 + d];
      const float phiA = sPhi[nl * 388 + d];
      const float phiB = sPhi[nl * 388 + 128 + d];
      const float phiC = sPhi[nl * 388 + 256 + d];
      const float fA = sF[nl * 52 + dd] + gE0 * bfa;            // m=0, j=d
      const float fC = sF[nl * 52 + 32 + dd] + gE0 * bfc;       // m=0, j=2D+d
      const size_t base = (size_t)n * 128 + d;
      dsc[base] = phiA * fA;
      const float cterm = phiC * fC;
      const float gEk[3] = {gE1, gE2, gE3};
#pragma unroll
      for (int k = 0; k < 3; ++k) {
        const float fB = sF[((k + 1) * 16 + nl) * 52 + 16 + dd] + gEk[k] * bfb;
        dv[base * 3 + k] = phiB * fB + cterm * vemb[base * 3 + k];
      }
    }
    __syncthreads();  // sF reused next phase
  }
}

// ---------------------------------------------------------------------------
extern "C" void kernel_launch(void* const* d_in, const int* in_sizes, int n_in,
                              void* d_out, int out_size, void* d_ws, size_t ws_size,
                              hipStream_t stream) {
  (void)in_sizes; (void)n_in; (void)out_size; (void)ws_size;
  const float* vemb = (const float*)d_in[0];   // vector_embeddings [N,128,3]
  const float* S    = (const float*)d_in[1];   // scalar_embeddings [N,128]
  const int*   nbr  = (const int*)d_in[2];     // neighbour_index [2,E]
  const float* ev   = (const float*)d_in[3];   // edge_vectors [E,3]
  const float* freq = (const float*)d_in[4];   // [20]
  const float* Wf   = (const float*)d_in[5];   // [20,384]
  const float* bf   = (const float*)d_in[6];   // [384]
  const float* W1   = (const float*)d_in[7];   // [128,128]
  const float* b1   = (const float*)d_in[8];   // [128]
  const float* W2   = (const float*)d_in[9];   // [128,384]
  const float* b2   = (const float*)d_in[10];  // [384]

  float* out = (float*)d_out;
  float* delta_v = out;                          // [N,128,3] first
  float* delta_s = out + (size_t)Nn * Dd * 3;    // [N,128] second

  float* G = (float*)d_ws;                       // [N,4,21] ~ 10.5 MB
  hipMemsetAsync(G, 0, (size_t)Nn * GROW * sizeof(float), stream);

  edge_accum_kernel<<<Ee / 256, 256, 0, stream>>>(nbr, ev, freq, G);
  node_fused_kernel<<<Nn / 16, 256, 0, stream>>>(S, W1, b1, W2, b2, Wf, bf, G,
                                                 vemb, delta_v, delta_s);
}